// ImprovedProteinPocketEncoder_21474836480096
// MI455X (gfx1250) — compile-verified
//
#include <hip/hip_runtime.h>
#include <hip/hip_bf16.h>
#include <stdint.h>

typedef float v2f __attribute__((ext_vector_type(2)));
typedef float v8f __attribute__((ext_vector_type(8)));

#define DIRECT_R 3.5f
#define EXT_R    8.0f
#define NEIGH_R2 25.0f
#define CHUNKS   8          // column-loop split for parallelism (2 row tiles/wave)

// ---------------------------------------------------------------------------
// Kernel 1: per-atom prep.
//   pkA[i]  = (x, y, z, 1)                      -> A rows (K=4)
//   pkB[j]  = (-2x, -2y, -2z, far? |p|^2 :1e30) -> B columns (bias kills !far)
//   nsq[i]  = |p|^2                             -> loop-invariant C init
//   counts zeroed for atomicAdd accumulation.
// ---------------------------------------------------------------------------
__global__ void prep_kernel(const float* __restrict__ pos,
                            const float* __restrict__ lig,
                            int N, int M,
                            float4* __restrict__ pkA,
                            float4* __restrict__ pkB,
                            float*  __restrict__ nsqArr,
                            float*  __restrict__ mind,
                            int*    __restrict__ farf,
                            int*    __restrict__ counts,
                            int*    __restrict__ maxn) {
    __shared__ float slig[768];                 // up to 256 ligand atoms
    const int t = threadIdx.x;
    for (int m = t; m < M * 3; m += blockDim.x) slig[m] = lig[m];
    __syncthreads();

    const int i = blockIdx.x * blockDim.x + t;
    if (i == 0) *maxn = 0;                      // re-init every call (graph replay safe)
    if (i < 16) {                               // zero the prefetch pad tile past pkB
        pkB[N + i] = make_float4(0.f, 0.f, 0.f, 0.f);
    }
    if (i >= N) return;

    const float x = pos[i * 3 + 0];
    const float y = pos[i * 3 + 1];
    const float z = pos[i * 3 + 2];

    float best = 3.4e38f;
    for (int m = 0; m < M; ++m) {
        const float dx = x - slig[m * 3 + 0];
        const float dy = y - slig[m * 3 + 1];
        const float dz = z - slig[m * 3 + 2];
        best = fminf(best, dx * dx + dy * dy + dz * dz);
    }
    const float md    = sqrtf(fmaxf(best, 1e-12f));
    const int   isfar = (md > EXT_R) ? 1 : 0;
    const float nsq   = x * x + y * y + z * z;

    pkA[i]    = make_float4(x, y, z, 1.0f);
    pkB[i]    = make_float4(-2.0f * x, -2.0f * y, -2.0f * z, isfar ? nsq : 1e30f);
    nsqArr[i] = nsq;
    mind[i]   = md;
    farf[i]   = isfar;
    counts[i] = 0;
}

// ---------------------------------------------------------------------------
// Kernel 2: N x N pairwise-distance counting via V_WMMA_F32_16X16X4_F32.
// One wave owns (32-row pair-tile, column chunk). Per iteration:
//   1x prefetched global_load_b64 (B half) + 2x v_wmma + 16x (v_cmp+carry add).
// Wave-uniform control scalars forced via readfirstlane -> scalar loop.
// C accumulator inits (|p_row|^2) are loop-invariant -> pinned in SRC2.
// Diagonal is intentionally counted (d2~0 < 25 iff row is far); corrected -1
// in the finalize pass. Partial row counts merged with atomicAdd.
// ---------------------------------------------------------------------------
__global__ void count_kernel(const float4* __restrict__ pkA,
                             const float4* __restrict__ pkB,
                             const float*  __restrict__ nsqArr,
                             int* __restrict__ counts,
                             int N) {
    const int lane   = threadIdx.x & 31;
    const int nTiles = N >> 4;
    const int nPairs = nTiles >> 1;

    // Wave-uniform scalars (readfirstlane so the loop control is SALU).
    const int wid    = __builtin_amdgcn_readfirstlane(threadIdx.x >> 5);
    const int wPerB  = blockDim.x >> 5;
    const int linear = __builtin_amdgcn_readfirstlane((int)blockIdx.x * wPerB + wid);
    if (linear >= nPairs * CHUNKS) return;      // uniform per wave (EXEC all-1s)

    const int pairTile = linear / CHUNKS;
    const int chunk    = linear - pairTile * CHUNKS;
    const int rowBase  = pairTile << 5;         // 32 rows per wave
    const int l15      = lane & 15;
    const int hi       = (lane >> 4) & 1;

    // Loop-invariant A operands: lanes 0-15 carry (K0,K1)=(x,y); 16-31 (K2,K3)=(z,1)
    const float* aBase = reinterpret_cast<const float*>(pkA);
    const v2f a0 = *reinterpret_cast<const v2f*>(aBase + ((rowBase      + l15) << 2) + (hi << 1));
    const v2f a1 = *reinterpret_cast<const v2f*>(aBase + ((rowBase + 16 + l15) << 2) + (hi << 1));

    // Loop-invariant C inits: c[v] = |p_row|^2 for row = base + hi*8 + v.
    float ci0[8], ci1[8];
#pragma unroll
    for (int v = 0; v < 8; ++v) {
        ci0[v] = nsqArr[rowBase      + (hi << 3) + v];
        ci1[v] = nsqArr[rowBase + 16 + (hi << 3) + v];
    }
    const v8f cinit0 = { ci0[0], ci0[1], ci0[2], ci0[3], ci0[4], ci0[5], ci0[6], ci0[7] };
    const v8f cinit1 = { ci1[0], ci1[1], ci1[2], ci1[3], ci1[4], ci1[5], ci1[6], ci1[7] };

    int cnt0[8], cnt1[8];
#pragma unroll
    for (int v = 0; v < 8; ++v) { cnt0[v] = 0; cnt1[v] = 0; }

    const int span    = nTiles / CHUNKS;
    const int ctBegin = chunk * span;

    // B pointer: per-iteration bump of 64 floats (one 16-column tile = 256 B).
    const float* bPtr = reinterpret_cast<const float*>(pkB)
                        + (ctBegin << 6) + (l15 << 2) + (hi << 1);

    v2f bCur = *reinterpret_cast<const v2f*>(bPtr);
    for (int it = 0; it < span; ++it) {
        bPtr += 64;
        // Prefetch next tile (last iteration over-reads into the zero pad tile).
        const v2f bNext = *reinterpret_cast<const v2f*>(bPtr);

        // d2 = |p_i|^2 + (-2 p_i.p_j + biased |p_j|^2)
        const v8f d0 = __builtin_amdgcn_wmma_f32_16x16x4_f32(
                false, a0, false, bCur, (short)0, cinit0, false, false);
        const v8f d1 = __builtin_amdgcn_wmma_f32_16x16x4_f32(
                false, a1, false, bCur, (short)0, cinit1, false, false);

#pragma unroll
        for (int v = 0; v < 8; ++v) {
            cnt0[v] += (d0[v] < NEIGH_R2) ? 1 : 0;   // v_cmp + carry-in add
            cnt1[v] += (d1[v] < NEIGH_R2) ? 1 : 0;
        }
        bCur = bNext;
    }

    // Reduce across the 16-lane half-wave sharing each row, merge via atomicAdd.
#pragma unroll
    for (int v = 0; v < 8; ++v) {
        int s0 = cnt0[v], s1 = cnt1[v];
        s0 += __shfl_xor(s0, 1); s1 += __shfl_xor(s1, 1);
        s0 += __shfl_xor(s0, 2); s1 += __shfl_xor(s1, 2);
        s0 += __shfl_xor(s0, 4); s1 += __shfl_xor(s1, 4);
        s0 += __shfl_xor(s0, 8); s1 += __shfl_xor(s1, 8);
        if (l15 == 0) {
            const int row = rowBase + (hi << 3) + v;
            atomicAdd(&counts[row], s0);
            atomicAdd(&counts[row + 16], s1);
        }
    }
}

// ---------------------------------------------------------------------------
// Kernel 3: finalize counts (remove self-count for far rows, zero others)
// and reduce max_n.
// ---------------------------------------------------------------------------
__global__ void finalize_kernel(const int* __restrict__ farf,
                                int* __restrict__ counts,
                                int* __restrict__ maxn,
                                int N) {
    const int i = blockIdx.x * blockDim.x + threadIdx.x;
    if (i >= N) return;
    const int val = farf[i] ? (counts[i] - 1) : 0;
    counts[i] = val;
    atomicMax(maxn, val);
}

// ---------------------------------------------------------------------------
// Kernel 4: tiered scores + chemical-property bonus table.
// ---------------------------------------------------------------------------
__global__ void score_kernel(const float* __restrict__ mind,
                             const int*   __restrict__ counts,
                             const int*   __restrict__ maxn,
                             const float* __restrict__ x,
                             int N, int F,
                             float* __restrict__ scores) {
    const int i = blockIdx.x * blockDim.x + threadIdx.x;
    if (i >= N) return;

    const float md = mind[i];
    float mn = (float)(*maxn);
    mn = (mn > 0.0f) ? mn : 1.0f;
    const float surface = 1.0f - (float)counts[i] / (mn + 1e-6f);

    float s;
    if (md <= DIRECT_R)      s = 10.0f;
    else if (md <= EXT_R)    s = 5.0f * (EXT_R - md) / EXT_R;
    else                     s = surface;

    int res = (int)x[i * F + 1];
    res = res < 0 ? 0 : (res > 19 ? 19 : res);
    static const float table[20] = {
        1.05f, 1.3f, 1.1f, 1.2f, 1.0f, 1.1f, 1.2f, 1.0f, 1.4f, 1.0f,
        1.0f,  1.3f, 1.0f, 1.5f, 1.0f, 1.05f, 1.05f, 1.6f, 1.4f, 1.1f };
    scores[i] = s * table[res];
}

// ---------------------------------------------------------------------------
// Kernel 5: top-K via single-workgroup bitonic sort of u64 sortable keys
// in dynamic LDS (N*8 = 128 KB; CDNA5 WGP has 320 KB).
// key = (ordered_float_bits(score) << 32) | ~index  -> descending sort gives
// exactly jax.lax.top_k order (value desc, index asc on ties).
// ---------------------------------------------------------------------------
__global__ void topk_kernel(const float* __restrict__ scores,
                            int N,
                            const int* __restrict__ maxAtomsPtr,
                            float* __restrict__ outBase) {
    extern __shared__ unsigned long long keys[];
    const int t  = threadIdx.x;
    const int nt = blockDim.x;
    const int K  = *maxAtomsPtr;

    for (int i = t; i < N; i += nt) {
        unsigned int b = __float_as_uint(scores[i]);
        b ^= (unsigned int)(((int)b >> 31)) | 0x80000000u;          // total-order map
        keys[i] = ((unsigned long long)b << 32) | (unsigned int)(~i);
    }
    __syncthreads();

    for (int k = 2; k <= N; k <<= 1) {
        for (int j = k >> 1; j > 0; j >>= 1) {
            for (int i = t; i < N; i += nt) {
                const int ixj = i ^ j;
                if (ixj > i) {
                    const unsigned long long a  = keys[i];
                    const unsigned long long bb = keys[ixj];
                    const bool descBlock = ((i & k) == 0);
                    if (descBlock ? (a < bb) : (a > bb)) {
                        keys[i]   = bb;
                        keys[ixj] = a;
                    }
                }
            }
            __syncthreads();
        }
    }

    float* topv = outBase + N;
    float* topi = topv + K;
    for (int i = t; i < K; i += nt) {
        const unsigned long long kk = keys[i];
        unsigned int b = (unsigned int)(kk >> 32);
        b ^= (unsigned int)(((int)(~b) >> 31)) | 0x80000000u;       // inverse map
        const unsigned int idx = ~((unsigned int)kk);
        topv[i] = __uint_as_float(b);
        topi[i] = (float)idx;                                       // indices as f32
    }
}

// ---------------------------------------------------------------------------
extern "C" void kernel_launch(void* const* d_in, const int* in_sizes, int n_in,
                              void* d_out, int out_size, void* d_ws, size_t ws_size,
                              hipStream_t stream) {
    const float* pos = (const float*)d_in[0];       // [N,3]
    const float* lig = (const float*)d_in[1];       // [M,3]
    const float* x   = (const float*)d_in[2];       // [N,F]
    const int*   mk  = (const int*)d_in[3];         // max_atoms (device scalar)

    const int N = in_sizes[0] / 3;
    const int M = in_sizes[1] / 3;
    const int F = in_sizes[2] / N;

    // Workspace carve-up (bytes). pkB has one extra 16-elem tile (256 B pad)
    // so the software-pipelined prefetch may safely over-read.
    char* ws = (char*)d_ws;
    float4* pkA    = (float4*)(ws);                                  // N*16
    float4* pkB    = (float4*)(ws + (size_t)N * 16);                 // N*16 + 256
    float*  nsqArr = (float*) (ws + (size_t)N * 32 + 256);           // N*4
    float*  mind   = (float*) (ws + (size_t)N * 36 + 256);           // N*4
    int*    farf   = (int*)   (ws + (size_t)N * 40 + 256);           // N*4
    int*    counts = (int*)   (ws + (size_t)N * 44 + 256);           // N*4
    int*    maxn   = (int*)   (ws + (size_t)N * 48 + 256);           // 4

    float* scores = (float*)d_out;

    // 1) prep
    prep_kernel<<<(N + 255) / 256, 256, 0, stream>>>(
        pos, lig, N, M, pkA, pkB, nsqArr, mind, farf, counts, maxn);

    // 2) N x N WMMA counting: one wave per (32-row pair-tile, column chunk).
    const int nTiles     = N / 16;
    const int totalWaves = (nTiles / 2) * CHUNKS;
    count_kernel<<<(totalWaves + 7) / 8, 256, 0, stream>>>(
        pkA, pkB, nsqArr, counts, N);

    // 3) finalize counts + max_n
    finalize_kernel<<<(N + 255) / 256, 256, 0, stream>>>(farf, counts, maxn, N);

    // 4) scores
    score_kernel<<<(N + 255) / 256, 256, 0, stream>>>(
        mind, counts, maxn, x, N, F, scores);

    // 5) top-K (N must be a power of two: 16384 here). 128 KB dynamic LDS.
    topk_kernel<<<1, 1024, (size_t)N * sizeof(unsigned long long), stream>>>(
        scores, N, mk, scores);
}